// dgmslice_ps_51668456571072
// MI455X (gfx1250) — compile-verified
//
#include <hip/hip_runtime.h>
#include <hip/hip_bf16.h>

typedef __attribute__((ext_vector_type(2))) float v2f;
typedef __attribute__((ext_vector_type(8))) float v8f;

// ---- fixed config (matches reference) ----
constexpr int kL = 32, kE = 1024, kV = 2048, kF = 8, kHU = 128, kHL = 2, kS = 7;
constexpr int kNS = kL * kE;          // 32768 samples through the MLP
constexpr int kD0 = 512, kD1R = 129, kD1E = 640;
constexpr int kP = 320;               // points per path (640 vals -> 320 pairs)
constexpr int kPADP = 512;            // bitonic pad
constexpr int kSIG = 584;             // 8 + 64 + 512
constexpr int kROW = 2 * kSIG;        // 1168
constexpr int kFINAL = kF * kROW;     // 9344
constexpr float kEPS = 1e-5f;

// -------------------------------------------------------------------------
// small utility kernels
// -------------------------------------------------------------------------
__global__ void zero_kernel(float* p, int n) {
  int i = blockIdx.x * blockDim.x + threadIdx.x;
  if (i < n) p[i] = 0.f;
}

// batch mean/var of the raw eigenvalues (shared by layer-1 of every MLP)
__global__ void x_stats_kernel(const float* __restrict__ x, float* __restrict__ st) {
  __shared__ float s1[256], s2[256];
  float a = 0.f, b = 0.f;
  for (int i = threadIdx.x; i < kNS; i += 256) { float v = x[i]; a += v; b += v * v; }
  s1[threadIdx.x] = a; s2[threadIdx.x] = b; __syncthreads();
  for (int o = 128; o > 0; o >>= 1) {
    if (threadIdx.x < o) { s1[threadIdx.x] += s1[threadIdx.x + o]; s2[threadIdx.x] += s2[threadIdx.x + o]; }
    __syncthreads();
  }
  if (threadIdx.x == 0) {
    float mx = s1[0] / kNS;
    st[0] = mx;
    st[1] = s2[0] / kNS - mx * mx;
  }
}

// layer 1: z = w_j * x  (bias cancels in BN); analytic batch stats
__global__ void layer1_kernel(const float* __restrict__ x, const float* __restrict__ win,
                              const float* __restrict__ g, const float* __restrict__ beta,
                              const float* __restrict__ xst, float* __restrict__ h) {
  int idx = blockIdx.x * 256 + threadIdx.x;
  if (idx >= kNS * kHU) return;
  int n = idx / kHU, j = idx - n * kHU;
  float w = win[j];
  float mx = xst[0], vx = xst[1];
  float v = (w * x[n] - w * mx) * rsqrtf(w * w * vx + kEPS) * g[j] + beta[j];
  h[idx] = fmaxf(v, 0.f);
}

// -------------------------------------------------------------------------
// hidden-layer GEMM: z[N,128] = h[N,128] @ W[128,128]^T  via f32 WMMA 16x16x4
// fused per-column sum / sumsq for BatchNorm statistics
// -------------------------------------------------------------------------
__global__ void gemm128_kernel(const float* __restrict__ h, const float* __restrict__ W,
                               float* __restrict__ z, float* __restrict__ colsum,
                               float* __restrict__ colsq) {
  __shared__ float At[16 * kHU];
  __shared__ float cs[kHU], cq[kHU];
  int tid = threadIdx.x;
  int row0 = blockIdx.x * 16;
  for (int i = tid; i < 16 * kHU; i += 256) At[i] = h[row0 * kHU + i];
  if (tid < kHU) { cs[tid] = 0.f; cq[tid] = 0.f; }
  __syncthreads();

  int wave = tid >> 5, lane = tid & 31;
  int n0 = wave * 16;                 // 8 waves cover all 128 output columns
  int m = lane & 15;
  int khalf = (lane >> 4) << 1;       // K pair: lanes 0-15 -> K0/K1, 16-31 -> K2/K3
  int ncol = n0 + m;
  v8f c = {};
  for (int kk = 0; kk < kHU; kk += 4) {
    v2f a, b;
    a.x = At[m * kHU + kk + khalf];
    a.y = At[m * kHU + kk + khalf + 1];
    const float* wp = &W[ncol * kHU + kk + khalf];   // B[k,n] = W[n,k]
    b.x = wp[0]; b.y = wp[1];
    c = __builtin_amdgcn_wmma_f32_16x16x4_f32(false, a, false, b, (short)0, c, false, false);
  }
  // C/D layout: VGPR i -> (M=i, N=lane) lanes 0-15 ; (M=8+i, N=lane-16) lanes 16-31
  int mbase = (lane < 16) ? 0 : 8;
  int col = n0 + (lane & 15);
  float ls = 0.f, lq = 0.f;
#pragma unroll
  for (int i = 0; i < 8; i++) {
    float v = c[i];
    z[(row0 + mbase + i) * kHU + col] = v;
    ls += v; lq += v * v;
  }
  atomicAdd(&cs[col], ls);
  atomicAdd(&cq[col], lq);
  __syncthreads();
  if (tid < kHU) { atomicAdd(&colsum[tid], cs[tid]); atomicAdd(&colsq[tid], cq[tid]); }
}

__global__ void finalize_stats_kernel(const float* __restrict__ cs, const float* __restrict__ cq,
                                      float* __restrict__ mean, float* __restrict__ inv, int m) {
  int j = blockIdx.x * blockDim.x + threadIdx.x;
  if (j < m) {
    float mu = cs[j] / kNS;
    float var = cq[j] / kNS - mu * mu;
    mean[j] = mu;
    inv[j] = rsqrtf(var + kEPS);
  }
}

__global__ void bn_relu_kernel(const float* __restrict__ z, const float* __restrict__ mean,
                               const float* __restrict__ inv, const float* __restrict__ g,
                               const float* __restrict__ beta, float* __restrict__ out) {
  int idx = blockIdx.x * 256 + threadIdx.x;
  if (idx >= kNS * kHU) return;
  int j = idx & (kHU - 1);
  float v = (z[idx] - mean[j]) * inv[j] * g[j] + beta[j];
  out[idx] = fmaxf(v, 0.f);
}

// output layer: z4[n] = <h[n,:], Wout>; fused scalar batch stats
__global__ void dot_out_kernel(const float* __restrict__ h, const float* __restrict__ wout,
                               float* __restrict__ zout, float* __restrict__ ssum) {
  __shared__ float sw[kHU];
  __shared__ float rs[256], rq[256];
  int tid = threadIdx.x;
  if (tid < kHU) sw[tid] = wout[tid];
  __syncthreads();
  int n = blockIdx.x * 256 + tid;
  float acc = 0.f;
  for (int j = 0; j < kHU; j++) acc += h[n * kHU + j] * sw[j];
  zout[n] = acc;
  rs[tid] = acc; rq[tid] = acc * acc; __syncthreads();
  for (int o = 128; o > 0; o >>= 1) {
    if (tid < o) { rs[tid] += rs[tid + o]; rq[tid] += rq[tid + o]; }
    __syncthreads();
  }
  if (tid == 0) { atomicAdd(&ssum[0], rs[0]); atomicAdd(&ssum[1], rq[0]); }
}

__global__ void scalar_bn_kernel(const float* __restrict__ z, const float* __restrict__ ssum,
                                 const float* __restrict__ g, const float* __restrict__ beta,
                                 float* __restrict__ out) {
  int n = blockIdx.x * 256 + threadIdx.x;
  if (n >= kNS) return;
  float mu = ssum[0] / kNS;
  float var = ssum[1] / kNS - mu * mu;
  float v = (z[n] - mu) * rsqrtf(var + kEPS) * g[0] + beta[0];
  out[n] = fmaxf(v, 0.f);
}

// -------------------------------------------------------------------------
// fv[l,f,v] = sum_e evs[f, l*E+e] * evsq[l,e,v]  — [8x1024]@[1024x2048] per l
// M padded 8->16 with zeros; f32 WMMA 16x16x4
// -------------------------------------------------------------------------
__global__ void einsum_wmma_kernel(const float* __restrict__ evs, const float* __restrict__ evsq,
                                   float* __restrict__ fv) {
  int l = blockIdx.y;
  int wave = threadIdx.x >> 5, lane = threadIdx.x & 31;
  int v0 = (blockIdx.x * 8 + wave) * 16;
  int m = lane & 15;
  int khalf = (lane >> 4) << 1;
  int n = v0 + m;
  const float* arow = &evs[(size_t)m * kNS + (size_t)l * kE];   // valid only for m<8
  v8f c = {};
  for (int e = 0; e < kE; e += 4) {
    int k = e + khalf;
    v2f a, b;
    a.x = (m < 8) ? arow[k] : 0.f;
    a.y = (m < 8) ? arow[k + 1] : 0.f;
    const float* bp = &evsq[((size_t)l * kE + k) * kV + n];
    b.x = bp[0];
    b.y = bp[kV];
    c = __builtin_amdgcn_wmma_f32_16x16x4_f32(false, a, false, b, (short)0, c, false, false);
  }
  if (lane < 16) {
#pragma unroll
    for (int i = 0; i < 8; i++)                          // rows M=0..7 are the real f's
      fv[((size_t)(l * kF + i)) * kV + v0 + lane] = c[i];
  }
}

// -------------------------------------------------------------------------
// per-diagram: gather -> 7-slice projection -> sort -> level-3 signature
// one workgroup (128 threads) per diagram; state lives in LDS
// -------------------------------------------------------------------------
__global__ void sig_kernel(const float* __restrict__ fv,
                           const int* __restrict__ dgm0, const int* __restrict__ dgm1rel,
                           const int* __restrict__ dgm1ext,
                           const float* __restrict__ Wproj, const float* __restrict__ bproj,
                           float* __restrict__ xsig) {
  __shared__ float vals[kD1E];
  __shared__ float proj[kS][kPADP];
  __shared__ float s1[8], s2[64], s3[512];
  __shared__ float dxs[8];
  int tid = threadIdx.x;          // 128 threads
  int diag = blockIdx.x;          // 512 diagrams
  int row = diag >> 1;            // l*F + f
  int which = diag & 1;           // 0 -> Dgm0, 1 -> Dgm1
  const float* fr = &fv[(size_t)row * kV];

  if (which == 0) {
    for (int j = tid; j < kD0; j += 128) vals[j] = fr[dgm0[row * kD0 + j]];
    // rel[:,1:][::-1] : j -> column (128 - j)
    if (tid < 128) vals[kD0 + tid] = fr[dgm1rel[row * kD1R + (128 - tid)]];
  } else {
    for (int j = tid; j < kD1E; j += 128) vals[j] = fr[dgm1ext[row * kD1E + j]];
  }
  __syncthreads();

  // pairwise projection onto 7 slices, pad to 512 with +inf for ascending sort
  for (int idx = tid; idx < kS * kPADP; idx += 128) {
    int s = idx / kPADP, p = idx - s * kPADP;
    float v;
    if (p < kP)
      v = Wproj[s * 2] * vals[2 * p] + Wproj[s * 2 + 1] * vals[2 * p + 1] + bproj[s];
    else
      v = __builtin_inff();
    proj[s][p] = v;
  }
  __syncthreads();

  // bitonic sort each slice ascending
  for (int s = 0; s < kS; s++) {
    float* d = proj[s];
    for (int k = 2; k <= kPADP; k <<= 1) {
      for (int j = k >> 1; j > 0; j >>= 1) {
        for (int i = tid; i < kPADP; i += 128) {
          int l2 = i ^ j;
          if (l2 > i) {
            float a = d[i], b = d[l2];
            bool up = ((i & k) == 0);
            if (up ? (a > b) : (a < b)) { d[i] = b; d[l2] = a; }
          }
        }
        __syncthreads();
      }
    }
  }

  // level-3 signature via Chen's identity
  const float dt = 1.0f / (kP * (kP + 1) / 2.0f);
  if (tid < 8)
    dxs[tid] = (tid == 0) ? dt : (proj[tid - 1][1] - proj[tid - 1][0]);
  __syncthreads();
  if (tid < 8)  s1[tid] = dxs[tid];
  if (tid < 64) s2[tid] = 0.5f * dxs[tid >> 3] * dxs[tid & 7];
  {
#pragma unroll
    for (int q = 0; q < 4; q++) {
      int e = tid * 4 + q;
      s3[e] = dxs[e >> 6] * dxs[(e >> 3) & 7] * dxs[e & 7] * (1.0f / 6.0f);
    }
  }
  __syncthreads();

  for (int p = 1; p < kP - 1; p++) {     // dX[1..318]
    if (tid < 8)
      dxs[tid] = (tid == 0) ? dt : (proj[tid - 1][p + 1] - proj[tid - 1][p]);
    __syncthreads();
    float ns3[4], ns2 = 0.f, ns1 = 0.f;
#pragma unroll
    for (int q = 0; q < 4; q++) {
      int e = tid * 4 + q;
      int i = e >> 6, j2 = (e >> 3) & 7, k2 = e & 7;
      float di = dxs[i], dj = dxs[j2], dk = dxs[k2];
      ns3[q] = s3[e] + di * dj * dk * (1.0f / 6.0f) + 0.5f * s1[i] * dj * dk
             + s2[(i << 3) | j2] * dk;
    }
    if (tid < 64) {
      int i = tid >> 3, j2 = tid & 7;
      ns2 = s2[tid] + 0.5f * dxs[i] * dxs[j2] + s1[i] * dxs[j2];
    }
    if (tid < 8) ns1 = s1[tid] + dxs[tid];
    __syncthreads();
#pragma unroll
    for (int q = 0; q < 4; q++) s3[tid * 4 + q] = ns3[q];
    if (tid < 64) s2[tid] = ns2;
    if (tid < 8)  s1[tid] = ns1;
    __syncthreads();
  }

  float* outp = &xsig[(size_t)row * kROW + which * kSIG];
  if (tid < 8)  outp[tid] = s1[tid];
  if (tid < 64) outp[8 + tid] = s2[tid];
#pragma unroll
  for (int q = 0; q < 4; q++) outp[72 + tid * 4 + q] = s3[tid * 4 + q];
}

// -------------------------------------------------------------------------
// final linear + batch BN over 32 rows
// -------------------------------------------------------------------------
__global__ void final_dot_kernel(const float* __restrict__ xsig, const float* __restrict__ wfin,
                                 const float* __restrict__ bfin, float* __restrict__ y) {
  __shared__ float red[256];
  int l = blockIdx.x, tid = threadIdx.x;
  float acc = 0.f;
  for (int i = tid; i < kFINAL; i += 256) acc += xsig[(size_t)l * kFINAL + i] * wfin[i];
  red[tid] = acc; __syncthreads();
  for (int o = 128; o > 0; o >>= 1) {
    if (tid < o) red[tid] += red[tid + o];
    __syncthreads();
  }
  if (tid == 0) y[l] = red[0] + bfin[0];
}

__global__ void final_bn_kernel(const float* __restrict__ y, const float* __restrict__ g,
                                const float* __restrict__ beta, float* __restrict__ out) {
  __shared__ float sy[kL];
  int tid = threadIdx.x;  // 32
  sy[tid] = y[tid]; __syncthreads();
  float mu = 0.f, sq = 0.f;
  for (int i = 0; i < kL; i++) { mu += sy[i]; sq += sy[i] * sy[i]; }
  mu /= (float)kL;
  float var = sq / (float)kL - mu * mu;
  out[tid] = (sy[tid] - mu) * rsqrtf(var + kEPS) * g[0] + beta[0];
}

// -------------------------------------------------------------------------
extern "C" void kernel_launch(void* const* d_in, const int* in_sizes, int n_in,
                              void* d_out, int out_size, void* d_ws, size_t ws_size,
                              hipStream_t stream) {
  const float* eigenvalues = (const float*)d_in[0];
  const float* evsq        = (const float*)d_in[1];
  const int*   dgm0        = (const int*)d_in[2];
  const int*   dgm1rel     = (const int*)d_in[3];
  const int*   dgm1ext     = (const int*)d_in[4];
  const float* mlp_Win     = (const float*)d_in[5];
  // d_in[6] mlp_bin  : bias cancels inside training-mode BN
  const float* mlp_gin     = (const float*)d_in[7];
  const float* mlp_betain  = (const float*)d_in[8];
  const float* mlp_Wh      = (const float*)d_in[9];
  // d_in[10] mlp_bh : cancels
  const float* mlp_gh      = (const float*)d_in[11];
  const float* mlp_betah   = (const float*)d_in[12];
  const float* mlp_Wout    = (const float*)d_in[13];
  // d_in[14] mlp_bout : cancels
  const float* mlp_gout    = (const float*)d_in[15];
  const float* mlp_betaout = (const float*)d_in[16];
  const float* Wproj       = (const float*)d_in[17];
  const float* bproj       = (const float*)d_in[18];
  const float* Wfin        = (const float*)d_in[19];
  const float* bfin        = (const float*)d_in[20];
  const float* gfin        = (const float*)d_in[21];
  const float* betafin     = (const float*)d_in[22];
  (void)in_sizes; (void)n_in; (void)out_size; (void)ws_size;

  float* ws = (float*)d_ws;
  float* hA     = ws; ws += (size_t)kNS * kHU;     // 4.19 M
  float* hB     = ws; ws += (size_t)kNS * kHU;     // 4.19 M
  float* evs    = ws; ws += (size_t)kF * kNS;      // 256 K
  float* fv     = ws; ws += (size_t)kL * kF * kV;  // 512 K
  float* xsig   = ws; ws += (size_t)kL * kFINAL;   // 299 K
  float* xstats = ws; ws += 2;
  float* colsum = ws; ws += kHU;                   // colsum/colsq contiguous
  float* colsq  = ws; ws += kHU;
  float* meanv  = ws; ws += kHU;
  float* invv   = ws; ws += kHU;
  float* ssum   = ws; ws += 2;
  float* y      = ws; ws += kL;

  x_stats_kernel<<<1, 256, 0, stream>>>(eigenvalues, xstats);

  const int ew_blocks = (kNS * kHU) / 256;   // 16384
  for (int f = 0; f < kF; f++) {
    layer1_kernel<<<ew_blocks, 256, 0, stream>>>(
        eigenvalues, mlp_Win + f * kHU, mlp_gin + f * kHU, mlp_betain + f * kHU, xstats, hA);
    for (int l = 0; l < kHL; l++) {
      zero_kernel<<<1, 256, 0, stream>>>(colsum, 2 * kHU);
      gemm128_kernel<<<kNS / 16, 256, 0, stream>>>(
          hA, mlp_Wh + (size_t)(f * kHL + l) * kHU * kHU, hB, colsum, colsq);
      finalize_stats_kernel<<<1, 128, 0, stream>>>(colsum, colsq, meanv, invv, kHU);
      bn_relu_kernel<<<ew_blocks, 256, 0, stream>>>(
          hB, meanv, invv, mlp_gh + (f * kHL + l) * kHU, mlp_betah + (f * kHL + l) * kHU, hA);
    }
    zero_kernel<<<1, 64, 0, stream>>>(ssum, 2);
    dot_out_kernel<<<kNS / 256, 256, 0, stream>>>(hA, mlp_Wout + f * kHU, evs + (size_t)f * kNS, ssum);
    scalar_bn_kernel<<<kNS / 256, 256, 0, stream>>>(
        evs + (size_t)f * kNS, ssum, mlp_gout + f, mlp_betaout + f, evs + (size_t)f * kNS);
  }

  einsum_wmma_kernel<<<dim3(kV / 128, kL), 256, 0, stream>>>(evs, evsq, fv);
  sig_kernel<<<2 * kL * kF, 128, 0, stream>>>(fv, dgm0, dgm1rel, dgm1ext, Wproj, bproj, xsig);
  final_dot_kernel<<<kL, 256, 0, stream>>>(xsig, Wfin, bfin, y);
  final_bn_kernel<<<1, kL, 0, stream>>>(y, gfin, betafin, (float*)d_out);
}